// RecursiveLogit_73985106641248
// MI455X (gfx1250) — compile-verified
//
#include <hip/hip_runtime.h>

typedef __attribute__((ext_vector_type(2))) float v2f;
typedef __attribute__((ext_vector_type(8))) float v8f;

#define NEG (-1000000000.0f)

// ---------------------------------------------------------------------------
// util = clip(feats @ weight.T + bias, -100, -1e-6)
// One wave32 computes 16 rows: D(16x16) = A(16x4,f32) x B(4x16,f32) chained
// over K=64 (16x v_wmma_f32_16x16x4_f32). B's 16 columns all hold the same
// weight chunk, so every column of D is the desired dot product.
// A layout (ISA 7.12.2, 32-bit A 16x4): lanes 0-15 -> M=lane, VGPR0=K0,VGPR1=K1;
// lanes 16-31 -> M=lane-16, VGPR0=K2, VGPR1=K3.  B mirrors with N across lanes.
// ---------------------------------------------------------------------------
__global__ void util_wmma_kernel(const float* __restrict__ feats,
                                 const float* __restrict__ weight,
                                 const float* __restrict__ bias,
                                 float* __restrict__ util, int E) {
    __shared__ float wl[64];
    if (threadIdx.x < 64) wl[threadIdx.x] = weight[threadIdx.x];
    __syncthreads();

    const int wave = threadIdx.x >> 5;
    const int lane = threadIdx.x & 31;
    const long long base = (long long)blockIdx.x * 128 + (long long)wave * 16;
    if (base >= E) return;

    const float b0 = bias[0];

    if (base + 16 <= E) {
        const int half = lane >> 4;       // 0: K0/K1, 1: K2/K3
        const int m    = lane & 15;       // row within tile
        const int koff = half * 2;
        const float* rowp = feats + (base + m) * 64;

        // Prefetch the HBM stream ~64 rows (16 KB) ahead into L2.
        __builtin_prefetch(rowp + 64 * 64, 0, 1);

        v8f c = {};
#pragma unroll
        for (int k0 = 0; k0 < 64; k0 += 4) {
            v2f a = *(const v2f*)(rowp + k0 + koff);          // feats tile
            v2f b = *(const v2f*)(&wl[k0 + koff]);            // weight bcast
            c = __builtin_amdgcn_wmma_f32_16x16x4_f32(
                    /*neg_a=*/false, a, /*neg_b=*/false, b,
                    /*c_mod=*/(short)0, c, /*reuse_a=*/false, /*reuse_b=*/false);
        }

        // D layout: VGPR r -> M=r (lanes 0-15) / M=8+r (lanes 16-31); all N
        // columns identical. Lane r stores row r, lane 16+r stores row 8+r.
#pragma unroll
        for (int r = 0; r < 8; ++r) {
            float uo = fminf(fmaxf(c[r] + b0, -100.0f), -1e-6f);
            if (lane == r)      util[base + r]     = uo;
            if (lane == 16 + r) util[base + 8 + r] = uo;
        }
    } else {
        // Tail (not hit for E = 1.6M, kept for generality).
        if (lane < 16 && base + lane < E) {
            const float* rowp = feats + (base + lane) * 64;
            float acc = 0.0f;
            for (int k = 0; k < 64; ++k) acc += rowp[k] * wl[k];
            util[base + lane] = fminf(fmaxf(acc + b0, -100.0f), -1e-6f);
        }
    }
}

// v0 = where(dest_mask, 0, NEG); reset reduction buffers.
__global__ void init_kernel(const unsigned char* __restrict__ dest_mask,
                            float* __restrict__ value,
                            unsigned* __restrict__ m_bits,
                            float* __restrict__ s_arr, int N) {
    int n = blockIdx.x * blockDim.x + threadIdx.x;
    if (n >= N) return;
    value[n]  = dest_mask[n] ? 0.0f : NEG;
    m_bits[n] = 0xFF800000u;   // bits of -inf
    s_arr[n]  = 0.0f;
}

// Pass 1: segment max of x = u[e] + v[dst[e]] into m[src[e]].
// All x < 0, so unsigned atomicMin on the raw bits == float max.
__global__ void edge_max_kernel(const int* __restrict__ src,
                                const int* __restrict__ dst,
                                const float* __restrict__ u,
                                const float* __restrict__ v,
                                unsigned* __restrict__ m_bits, int E) {
    int e = blockIdx.x * blockDim.x + threadIdx.x;
    if (e >= E) return;
    float x = u[e] + v[dst[e]];
    atomicMin(&m_bits[src[e]], __float_as_uint(x));
}

// Pass 2: s[src[e]] += exp(x - max(m[src[e]], NEG))
__global__ void edge_sum_kernel(const int* __restrict__ src,
                                const int* __restrict__ dst,
                                const float* __restrict__ u,
                                const float* __restrict__ v,
                                const unsigned* __restrict__ m_bits,
                                float* __restrict__ s_arr, int E) {
    int e = blockIdx.x * blockDim.x + threadIdx.x;
    if (e >= E) return;
    int si = src[e];
    float x  = u[e] + v[dst[e]];
    float mc = fmaxf(__uint_as_float(m_bits[si]), NEG);
    atomicAdd(&s_arr[si], __expf(x - mc));
}

// v_new = where(dest_mask, 0, s>0 ? log(s)+m : NEG); reset m,s for next iter.
__global__ void node_update_kernel(const unsigned char* __restrict__ dest_mask,
                                   float* __restrict__ value,
                                   unsigned* __restrict__ m_bits,
                                   float* __restrict__ s_arr, int N) {
    int n = blockIdx.x * blockDim.x + threadIdx.x;
    if (n >= N) return;
    float mc = fmaxf(__uint_as_float(m_bits[n]), NEG);
    float sv = s_arr[n];
    float vn = (sv > 0.0f) ? (__logf(sv) + mc) : NEG;
    value[n]  = dest_mask[n] ? 0.0f : vn;
    m_bits[n] = 0xFF800000u;
    s_arr[n]  = 0.0f;
}

// prob = exp(u + value[dst] - value[src])
__global__ void prob_kernel(const int* __restrict__ src,
                            const int* __restrict__ dst,
                            const float* __restrict__ u,
                            const float* __restrict__ value,
                            float* __restrict__ prob, int E) {
    int e = blockIdx.x * blockDim.x + threadIdx.x;
    if (e >= E) return;
    prob[e] = __expf(u[e] + value[dst[e]] - value[src[e]]);
}

extern "C" void kernel_launch(void* const* d_in, const int* in_sizes, int n_in,
                              void* d_out, int out_size, void* d_ws, size_t ws_size,
                              hipStream_t stream) {
    const float*         feats     = (const float*)d_in[0];
    const unsigned char* dest_mask = (const unsigned char*)d_in[1];
    const int*           edge_idx  = (const int*)d_in[2];
    // d_in[3] = batch (unused), d_in[4] = n_nodes scalar (host sizes suffice)
    const float*         weight    = (const float*)d_in[5];
    const float*         bias      = (const float*)d_in[6];

    const int N = in_sizes[1];        // 50000
    const int E = in_sizes[2] / 2;    // 1600000
    const int* src = edge_idx;
    const int* dst = edge_idx + E;

    float* value = (float*)d_out;     // N  (iterated in place -> final output)
    float* util  = value + N;         // E
    float* prob  = util + E;          // E

    unsigned* m_bits = (unsigned*)d_ws;       // N u32
    float*    s_arr  = (float*)d_ws + N;      // N f32

    const dim3 blk(256);
    const int gridE = (E + 255) / 256;
    const int gridN = (N + 255) / 256;
    const int gridU = (int)((E + 127) / 128);

    util_wmma_kernel<<<gridU, blk, 0, stream>>>(feats, weight, bias, util, E);
    init_kernel<<<gridN, blk, 0, stream>>>(dest_mask, value, m_bits, s_arr, N);

    for (int it = 0; it < 32; ++it) {
        edge_max_kernel<<<gridE, blk, 0, stream>>>(src, dst, util, value, m_bits, E);
        edge_sum_kernel<<<gridE, blk, 0, stream>>>(src, dst, util, value, m_bits, s_arr, E);
        node_update_kernel<<<gridN, blk, 0, stream>>>(dest_mask, value, m_bits, s_arr, N);
    }

    prob_kernel<<<gridE, blk, 0, stream>>>(src, dst, util, value, prob, E);
}